// LatentCapsLayer_3521873183191
// MI455X (gfx1250) — compile-verified
//
#include <hip/hip_runtime.h>
#include <math.h>

typedef float v2f __attribute__((ext_vector_type(2)));
typedef float v8f __attribute__((ext_vector_type(8)));
typedef int   v4i __attribute__((ext_vector_type(4)));

// address-space-qualified pointer types for the async-copy builtin
typedef __attribute__((address_space(1))) v4i AS1_v4i;
typedef __attribute__((address_space(3))) v4i AS3_v4i;

#define B_ 16
#define L_ 64
#define P_ 1024
#define V_ 64
#define Q_ 16
#define CHUNK 4          // p's per block (even: stage-buffer parity == pi&1)
#define THREADS 256      // 8 waves (wave32)
#define WAVES 8
#define L_PER_WAVE 8
#define LN64 4.158883083359672f

#if defined(__HIP_PLATFORM_AMD__)
#define FATOMIC_ADD(p, v) unsafeAtomicAdd((p), (v))
#else
#define FATOMIC_ADD(p, v) atomicAdd((p), (v))
#endif

// ---- CDNA5 async global->LDS copy path (guarded; falls back to direct loads) ----
#if __has_builtin(__builtin_amdgcn_global_load_async_to_lds_b128)
#define HAVE_ASYNC_COPY 1
#else
#define HAVE_ASYNC_COPY 0
#endif

#define WAIT_ASYNC(n) asm volatile("s_wait_asynccnt %0" ::"i"(n) : "memory")

// butterfly sum within each 16-lane half (wave32) via ds_swizzle immediates
#if __has_builtin(__builtin_amdgcn_ds_swizzle)
__device__ __forceinline__ float half16_sum(float v) {
    v += __int_as_float(__builtin_amdgcn_ds_swizzle(__float_as_int(v), 0x041F)); // xor 1
    v += __int_as_float(__builtin_amdgcn_ds_swizzle(__float_as_int(v), 0x081F)); // xor 2
    v += __int_as_float(__builtin_amdgcn_ds_swizzle(__float_as_int(v), 0x101F)); // xor 4
    v += __int_as_float(__builtin_amdgcn_ds_swizzle(__float_as_int(v), 0x201F)); // xor 8
    return v;
}
#else
__device__ __forceinline__ float half16_sum(float v) {
    v += __shfl_xor(v, 1);
    v += __shfl_xor(v, 2);
    v += __shfl_xor(v, 4);
    v += __shfl_xor(v, 8);
    return v;
}
#endif

// Dynamic LDS layout (floats):
//   [0, 4096)            c_lds  : coupling coeffs / logits  (CHUNK*B_*L_)
//   [4096, 4160)         Tpart
//   [4160, 4160+16384)   per-wave double-buffered W-tile stage: 8 waves x 2 x 1024
#define C_OFF   0
#define T_OFF   4096
#define STG_OFF 4160
#define SMEM_FLOATS (STG_OFF + WAVES * 2 * 1024)
#define SMEM_BYTES  (SMEM_FLOATS * 4)

// Stage one 4KB W tile (64 v x 16 q floats) into LDS: 8 x b128 per lane, coalesced.
__device__ __forceinline__ void tile_prefetch(const float* wlp, float* stage, int lane) {
#if HAVE_ASYNC_COPY
#pragma unroll
    for (int i = 0; i < 8; ++i) {
        __builtin_amdgcn_global_load_async_to_lds_b128(
            (AS1_v4i*)(wlp + lane * 4 + i * 128),
            (AS3_v4i*)(stage + lane * 4 + i * 128),
            0, 0);
    }
#else
    (void)wlp; (void)stage; (void)lane;
#endif
}

// One routing pass. it=0: uniform c=1/64. it>0: logits = <vsum, u_hat>, softmax over l.
// Accumulates s[b,l,v] (+= c * u_hat) via f32 atomics; it==2 also accumulates T numerator.
__global__ __launch_bounds__(THREADS) void caps_pass_kernel(
    const float* __restrict__ W, const float* __restrict__ x,
    const float* __restrict__ vsum, float* __restrict__ s_out,
    float* __restrict__ T_out, int it)
{
    extern __shared__ __align__(16) float smem[];
    float* c_lds = smem + C_OFF;
    float* Tpart = smem + T_OFF;

    const int tid  = threadIdx.x;
    const int wave = tid >> 5;
    const int lane = tid & 31;
    const int lo   = lane & 15;     // A: m=b; B/C: n index
    const int hi   = lane >> 4;     // K offset (A/B), M offset (C)
    const int p0   = blockIdx.x * CHUNK;
    const int lbase = wave * L_PER_WAVE;

    float* stg = smem + STG_OFF + wave * 2048;  // two 1024-float buffers

    // W tile base for (l, pi)
#define WTILE(l_, pi_) (W + (size_t)(((l_) * P_ + (p0 + (pi_))) * V_) * Q_)

    // ---- A tiles (x) for the p-chunk: lane holds m=lo, VGPR j holds k=4*kc+2*hi+j ----
    v2f a[CHUNK][4];
#pragma unroll
    for (int pi = 0; pi < CHUNK; ++pi) {
        const float* xp = x + (lo * P_ + (p0 + pi)) * Q_;
#pragma unroll
        for (int kc = 0; kc < 4; ++kc)
            a[pi][kc] = *(const v2f*)(xp + 4 * kc + 2 * hi);
    }

    // ---------- Phase A: routing logits via WMMA ----------
    if (it > 0) {
#if HAVE_ASYNC_COPY
        tile_prefetch(WTILE(lbase, 0), stg, lane);
#endif
        for (int j = 0; j < L_PER_WAVE; ++j) {
            const int l = lbase + j;
            // vsum slice for this l in C-tile layout: vs[t][r] = vsum[r+8*hi][l][lo+16*t]
            float vs[4][8];
#pragma unroll
            for (int t = 0; t < 4; ++t)
#pragma unroll
                for (int r = 0; r < 8; ++r)
                    vs[t][r] = vsum[(((r + 8 * hi) * L_ + l) * V_) + lo + 16 * t];

#pragma unroll
            for (int pi = 0; pi < CHUNK; ++pi) {
#if HAVE_ASYNC_COPY
                if (pi < CHUNK - 1) {
                    tile_prefetch(WTILE(l, pi + 1), stg + ((pi + 1) & 1) * 1024, lane);
                    WAIT_ASYNC(8);
                } else if (j < L_PER_WAVE - 1) {
                    tile_prefetch(WTILE(l + 1, 0), stg, lane);  // pi2=0 -> buffer 0
                    WAIT_ASYNC(8);
                } else {
                    WAIT_ASYNC(0);
                }
                const float* tsrc = stg + (pi & 1) * 1024;
#else
                const float* tsrc = WTILE(l, pi);
#endif
                float part[8];
#pragma unroll
                for (int r = 0; r < 8; ++r) part[r] = 0.f;

#pragma unroll
                for (int vt = 0; vt < 4; ++vt) {
                    v8f c;
#pragma unroll
                    for (int r = 0; r < 8; ++r) c[r] = 0.f;
#pragma unroll
                    for (int kc = 0; kc < 4; ++kc) {
                        v2f b = *(const v2f*)(tsrc + (lo + 16 * vt) * Q_ + 4 * kc + 2 * hi);
                        c = __builtin_amdgcn_wmma_f32_16x16x4_f32(
                                false, a[pi][kc], false, b, (short)0, c, false, false);
                    }
#pragma unroll
                    for (int r = 0; r < 8; ++r) part[r] += c[r] * vs[vt][r];
                }
                // reduce over v within each 16-lane half; lanes with lo==0 hold result
#pragma unroll
                for (int r = 0; r < 8; ++r) {
                    float v = half16_sum(part[r]);
                    if (lo == 0)
                        c_lds[((pi * B_) + (r + 8 * hi)) * L_ + l] = v;
                }
            }
        }
    }
    __syncthreads();

    // ---------- softmax over l (or uniform on iteration 0) ----------
    if (it == 0) {
        for (int i = tid; i < CHUNK * B_ * L_; i += THREADS)
            c_lds[i] = 1.0f / 64.0f;
    } else if (tid < CHUNK * B_) {
        float* row = &c_lds[tid * L_];
        float m = row[0];
        for (int l = 1; l < L_; ++l) m = fmaxf(m, row[l]);
        float sum = 0.f;
        for (int l = 0; l < L_; ++l) { float e = __expf(row[l] - m); row[l] = e; sum += e; }
        float inv = 1.0f / sum;
        float ent = 0.f;
        for (int l = 0; l < L_; ++l) {
            float cc = row[l] * inv;
            row[l] = cc;
            ent -= cc * __logf(cc + 1e-12f);
        }
        if (it == 2) Tpart[tid] = 1.0f - ent * (1.0f / LN64);
    }
    __syncthreads();

    // ---------- Phase B: s += c * u_hat, fused into WMMA (scale A rows by c) ----------
    {
#if HAVE_ASYNC_COPY
        tile_prefetch(WTILE(lbase, 0), stg, lane);
#endif
        for (int j = 0; j < L_PER_WAVE; ++j) {
            const int l = lbase + j;
            v8f ct[4];
#pragma unroll
            for (int t = 0; t < 4; ++t)
#pragma unroll
                for (int r = 0; r < 8; ++r) ct[t][r] = 0.f;

#pragma unroll
            for (int pi = 0; pi < CHUNK; ++pi) {
#if HAVE_ASYNC_COPY
                if (pi < CHUNK - 1) {
                    tile_prefetch(WTILE(l, pi + 1), stg + ((pi + 1) & 1) * 1024, lane);
                    WAIT_ASYNC(8);
                } else if (j < L_PER_WAVE - 1) {
                    tile_prefetch(WTILE(l + 1, 0), stg, lane);
                    WAIT_ASYNC(8);
                } else {
                    WAIT_ASYNC(0);
                }
                const float* tsrc = stg + (pi & 1) * 1024;
#else
                const float* tsrc = WTILE(l, pi);
#endif
                const float sc = c_lds[((pi * B_) + lo) * L_ + l];  // c[b=lo, l, p]
                v2f as[4];
#pragma unroll
                for (int kc = 0; kc < 4; ++kc) as[kc] = a[pi][kc] * sc;
#pragma unroll
                for (int vt = 0; vt < 4; ++vt) {
#pragma unroll
                    for (int kc = 0; kc < 4; ++kc) {
                        v2f b = *(const v2f*)(tsrc + (lo + 16 * vt) * Q_ + 4 * kc + 2 * hi);
                        ct[vt] = __builtin_amdgcn_wmma_f32_16x16x4_f32(
                                     false, as[kc], false, b, (short)0, ct[vt], false, false);
                    }
                }
            }
            // flush this wave's (l) contribution: s_out[b][l][v]
#pragma unroll
            for (int t = 0; t < 4; ++t)
#pragma unroll
                for (int r = 0; r < 8; ++r) {
                    const int b = r + 8 * hi;
                    const int v = lo + 16 * t;
                    FATOMIC_ADD(&s_out[((b * L_) + l) * V_ + v], ct[t][r]);
                }
        }
    }

    if (it == 2) {
        __syncthreads();
        if (tid == 0) {
            float tsum = 0.f;
            for (int i = 0; i < CHUNK * B_; ++i) tsum += Tpart[i];
            FATOMIC_ADD(T_out, tsum);
        }
    }
#undef WTILE
}

__global__ void caps_init_kernel(float* ws) {
    const int n = 2 * (B_ * L_ * V_) + 1;  // s, vsum, T accumulator
    for (int i = blockIdx.x * blockDim.x + threadIdx.x; i < n;
         i += gridDim.x * blockDim.x)
        ws[i] = 0.f;
}

// squash(s): scale = sqrt(|s|^2)/(1+|s|^2). mode 0: vsum += v, reset s.
// mode 1: write v to d_out and ||v|| = sq/(1+sq) to norms.
__global__ void caps_squash_kernel(float* __restrict__ s, float* __restrict__ vsum_buf,
                                   float* __restrict__ out_v, float* __restrict__ norms,
                                   int final_mode) {
    const int t = blockIdx.x * blockDim.x + threadIdx.x;  // (b*L + l)
    if (t >= B_ * L_) return;
    float* sp = s + t * V_;
    float vals[V_];
    float sq = 0.f;
#pragma unroll
    for (int v = 0; v < V_; ++v) { vals[v] = sp[v]; sq += vals[v] * vals[v]; }
    const float scale = sq / ((1.0f + sq) * sqrtf(sq));
    if (!final_mode) {
        float* vp = vsum_buf + t * V_;
#pragma unroll
        for (int v = 0; v < V_; ++v) { vp[v] += scale * vals[v]; sp[v] = 0.f; }
    } else {
#pragma unroll
        for (int v = 0; v < V_; ++v) out_v[t * V_ + v] = scale * vals[v];
        norms[t] = sq / (1.0f + sq);
    }
}

__global__ void caps_finish_kernel(const float* __restrict__ norms,
                                   const float* __restrict__ T_acc,
                                   float* __restrict__ out) {
    __shared__ float red[L_];
    const int l = threadIdx.x;  // 64 threads
    float m1 = 0.f, m2 = 0.f;
    for (int b = 0; b < B_; ++b) {
        const float n = norms[b * L_ + l];
        m1 += n; m2 += n * n;
    }
    m1 *= (1.0f / 16.0f); m2 *= (1.0f / 16.0f);
    red[l] = sqrtf(fmaxf(m2 - m1 * m1, 0.f));
    __syncthreads();
    for (int s2 = 32; s2 > 0; s2 >>= 1) {
        if (l < s2) red[l] = fmaxf(red[l], red[l + s2]);
        __syncthreads();
    }
    if (l == 0) {
        out[B_ * L_ * V_]     = T_acc[0] * (1.0f / (float)(B_ * P_));  // T
        out[B_ * L_ * V_ + 1] = red[0];                                // D
    }
}

extern "C" void kernel_launch(void* const* d_in, const int* in_sizes, int n_in,
                              void* d_out, int out_size, void* d_ws, size_t ws_size,
                              hipStream_t stream) {
    (void)in_sizes; (void)n_in; (void)out_size; (void)ws_size;
    const float* x = (const float*)d_in[0];  // (B, P, Q)
    const float* W = (const float*)d_in[1];  // (L, P, V, Q)
    float* out = (float*)d_out;              // v (B,L,V) | T | D
    float* ws  = (float*)d_ws;

    float* s_glob = ws;                       // 65536 floats
    float* vsum   = ws + B_ * L_ * V_;        // 65536 floats
    float* T_acc  = ws + 2 * B_ * L_ * V_;    // 1 float
    float* norms  = T_acc + 1;                // 1024 floats

    caps_init_kernel<<<256, 256, 0, stream>>>(ws);
    for (int it = 0; it < 3; ++it) {
        caps_pass_kernel<<<P_ / CHUNK, THREADS, SMEM_BYTES, stream>>>(
            W, x, vsum, s_glob, T_acc, it);
        caps_squash_kernel<<<4, 256, 0, stream>>>(s_glob, vsum, out, norms,
                                                  it == 2 ? 1 : 0);
    }
    caps_finish_kernel<<<1, L_, 0, stream>>>(norms, T_acc, out);
}